// MultiplicativeCrossAttention_28845000360519
// MI455X (gfx1250) — compile-verified
//
#include <hip/hip_runtime.h>
#include <hip/hip_bf16.h>

// CDNA5 (gfx1250) fused multiplicative cross-attention.
// proj = Q @ W ; logits = proj @ KV^T (+mask) ; softmax ; O = P @ KV ; out = [O | Q]
// All three GEMMs use V_WMMA_F32_16X16X4_F32 (full f32 precision on the matrix pipe).

typedef __attribute__((ext_vector_type(2))) float v2f;
typedef __attribute__((ext_vector_type(8))) float v8f;

#define SDIM 1024      // S_enc == S_dec == Q_SIZE == KV2
#define OUTW 2048      // 2kv + q
#define NEG_INF_F (-1e9f)

__global__ __launch_bounds__(256)
void mca_fused_kernel(const float* __restrict__ kv,     // [B,1024,1024]
                      const int*   __restrict__ mask,   // [B,1024]
                      const float* __restrict__ query,  // [B,1024,1024]
                      const float* __restrict__ W,      // [1024,1024]
                      float* __restrict__ out)          // [B,1024,2048]
{
    extern __shared__ float lds[];
    float* R0 = lds;              // 16x1024 : query tile -> logits -> probs
    float* R1 = lds + 16 * SDIM;  // 16x1024 : proj tile

    const int b    = blockIdx.y;
    const int d0   = blockIdx.x * 16;
    const int tid  = threadIdx.x;
    const int wid  = tid >> 5;        // 8 waves
    const int lane = tid & 31;        // wave32
    const int hi   = lane >> 4;       // lane half (0/1)
    const int lo   = lane & 15;       // lane within half
    const int n0   = wid * 128;       // this wave's 128 output columns (8 WMMA tiles)

    const float* qb  = query + ((size_t)b * SDIM + d0) * SDIM;
    const float* kvb = kv + (size_t)b * SDIM * SDIM;

    // ---- stage 16x1024 query tile into R0 (contiguous block) -------------
    {
        const float4* src = reinterpret_cast<const float4*>(qb);
        float4*       dst = reinterpret_cast<float4*>(R0);
#pragma unroll
        for (int i = 0; i < 16; ++i) dst[tid + i * 256] = src[tid + i * 256];
    }
    // warm L2 with this batch's kv slab (global_prefetch_b8); address is valid.
    __builtin_prefetch(kvb + (size_t)tid * 256, 0, 1);
    __syncthreads();

    // ================= Phase 1: proj = Q_tile @ W ========================
    v8f acc[8];
#pragma unroll
    for (int t = 0; t < 8; ++t) acc[t] = {};

    for (int k = 0; k < SDIM; k += 4) {
        // A fragment (16x4): lane half selects k+0/1 vs k+2/3
        v2f a = *reinterpret_cast<const v2f*>(&R0[lo * SDIM + k + 2 * hi]);
        const float* Wk = W + (size_t)(k + 2 * hi) * SDIM;
#pragma unroll
        for (int t = 0; t < 8; ++t) {
            const int n = n0 + t * 16 + lo;
            v2f bb;
            bb.x = Wk[n];           // row k+2*hi
            bb.y = Wk[SDIM + n];    // row k+2*hi+1
            acc[t] = __builtin_amdgcn_wmma_f32_16x16x4_f32(
                false, a, false, bb, (short)0, acc[t], false, false);
        }
    }
    __syncthreads();              // done reading R0 (query)
    // write proj tile to R1 per C/D layout: row = v + 8*hi, col = n
#pragma unroll
    for (int t = 0; t < 8; ++t)
#pragma unroll
        for (int v = 0; v < 8; ++v)
            R1[(v + 8 * hi) * SDIM + n0 + t * 16 + lo] = acc[t][v];
    __syncthreads();

    // ============ Phase 2A: logits = proj @ KV^T  (+ mask) ===============
#pragma unroll
    for (int t = 0; t < 8; ++t) acc[t] = {};

    for (int k = 0; k < SDIM; k += 4) {
        v2f a = *reinterpret_cast<const v2f*>(&R1[lo * SDIM + k + 2 * hi]);
#pragma unroll
        for (int t = 0; t < 8; ++t) {
            const int e = n0 + t * 16 + lo;  // B col = encoder index
            // B(k,e) = kv[e,k] : contiguous 8B per lane (8-byte aligned)
            v2f bb = *reinterpret_cast<const v2f*>(&kvb[(size_t)e * SDIM + k + 2 * hi]);
            acc[t] = __builtin_amdgcn_wmma_f32_16x16x4_f32(
                false, a, false, bb, (short)0, acc[t], false, false);
        }
    }
    {
        const int* mb = mask + b * SDIM;
#pragma unroll
        for (int t = 0; t < 8; ++t) {
            const int e = n0 + t * 16 + lo;
            const bool valid = (mb[e] != 0);
#pragma unroll
            for (int v = 0; v < 8; ++v)
                R0[(v + 8 * hi) * SDIM + e] = valid ? acc[t][v] : NEG_INF_F;
        }
    }
    __syncthreads();

    // ================= Softmax over e (rows of R0) =======================
    // 8 waves x 2 rows; wave32 shuffle reductions.
#pragma unroll
    for (int rr = 0; rr < 2; ++rr) {
        float* row = R0 + (wid * 2 + rr) * SDIM;
        float m = -3.4e38f;
        for (int e = lane; e < SDIM; e += 32) m = fmaxf(m, row[e]);
#pragma unroll
        for (int off = 16; off > 0; off >>= 1) m = fmaxf(m, __shfl_xor(m, off, 32));
        float s = 0.0f;
        for (int e = lane; e < SDIM; e += 32) {
            float p = __expf(row[e] - m);
            row[e] = p;
            s += p;
        }
#pragma unroll
        for (int off = 16; off > 0; off >>= 1) s += __shfl_xor(s, off, 32);
        const float inv = 1.0f / s;
        for (int e = lane; e < SDIM; e += 32) row[e] *= inv;
    }
    __syncthreads();

    // ================= Phase 2B: O = P @ KV ==============================
#pragma unroll
    for (int t = 0; t < 8; ++t) acc[t] = {};

    for (int e = 0; e < SDIM; e += 4) {
        v2f a = *reinterpret_cast<const v2f*>(&R0[lo * SDIM + e + 2 * hi]);
        const float* kr = kvb + (size_t)(e + 2 * hi) * SDIM;
#pragma unroll
        for (int t = 0; t < 8; ++t) {
            const int n = n0 + t * 16 + lo;
            v2f bb;
            bb.x = kr[n];           // kv row e+2*hi (coalesced across lanes)
            bb.y = kr[SDIM + n];    // kv row e+2*hi+1
            acc[t] = __builtin_amdgcn_wmma_f32_16x16x4_f32(
                false, a, false, bb, (short)0, acc[t], false, false);
        }
    }

    // ================= Epilogue: out = [O | Q] ===========================
    float* ob = out + ((size_t)b * SDIM + d0) * OUTW;
#pragma unroll
    for (int t = 0; t < 8; ++t)
#pragma unroll
        for (int v = 0; v < 8; ++v)
            ob[(size_t)(v + 8 * hi) * OUTW + n0 + t * 16 + lo] = acc[t][v];

    // concat query (read straight from global; L2-hot)
    for (int i = tid; i < 16 * SDIM; i += 256) {
        const int r = i >> 10;
        const int c = i & 1023;
        ob[(size_t)r * OUTW + SDIM + c] = qb[r * SDIM + c];
    }
}

extern "C" void kernel_launch(void* const* d_in, const int* in_sizes, int n_in,
                              void* d_out, int out_size, void* d_ws, size_t ws_size,
                              hipStream_t stream) {
    const float* kv    = (const float*)d_in[0];   // keyvalues  [32,1024,1024] f32
    const int*   mask  = (const int*)d_in[1];     // mask       [32,1024]      i32
    const float* query = (const float*)d_in[2];   // query      [32,1024,1024] f32
    const float* W     = (const float*)d_in[3];   // W          [1024,1024]    f32
    float*       out   = (float*)d_out;           // [32,1024,2048] f32

    const size_t smem = 2ull * 16 * SDIM * sizeof(float);  // 128 KB dynamic LDS
    (void)hipFuncSetAttribute((const void*)mca_fused_kernel,
                              hipFuncAttributeMaxDynamicSharedMemorySize, (int)smem);

    dim3 grid(SDIM / 16, 32);   // (d-tiles, batch)
    dim3 block(256);            // 8 wave32 waves
    mca_fused_kernel<<<grid, block, smem, stream>>>(kv, mask, query, W, out);
}